// CapsuleLayer_73074573574502
// MI455X (gfx1250) — compile-verified
//
#include <hip/hip_runtime.h>
#include <math.h>

#define B_SZ  64
#define N_IN  2304
#define D_INN 8
#define J_CAPS 32
#define L_DIM  16
#define JL (J_CAPS * L_DIM)   // 512
#define EPSQ 1e-7f
#define CH 64                  // K-chunk staged in LDS per loop
#define ASTR 68                // Asub row stride (floats): bank-safe, 16B-aligned

typedef float v8f __attribute__((ext_vector_type(8)));
typedef float v2f __attribute__((ext_vector_type(2)));

// ---------------- prep: xs[b][n] = sum_i x[b,n,i] ----------------
__global__ __launch_bounds__(256) void prep_x(const float* __restrict__ x,
                                              float* __restrict__ xs) {
    int t = blockIdx.x * 256 + threadIdx.x;          // t = b*N_IN + n
    const float4* p = (const float4*)(x + (size_t)t * D_INN);
    float4 a = p[0], b = p[1];
    xs[t] = (a.x + a.y) + (a.z + a.w) + (b.x + b.y) + (b.z + b.w);
}

// ---------------- prep: WsJ[j][n][l] = sum_k W[n,j,k,l] ----------------
__global__ __launch_bounds__(256) void prep_w(const float* __restrict__ W,
                                              float* __restrict__ WsJ) {
    int t = blockIdx.x * 256 + threadIdx.x;          // t = n*32 + j
    int n = t >> 5, j = t & 31;
    const float4* p = (const float4*)(W + (size_t)t * (D_INN * L_DIM));
    float4 acc[4];
#pragma unroll
    for (int q = 0; q < 4; ++q) acc[q] = make_float4(0.f, 0.f, 0.f, 0.f);
#pragma unroll
    for (int k = 0; k < D_INN; ++k)
#pragma unroll
        for (int q = 0; q < 4; ++q) {
            float4 w = p[k * 4 + q];
            acc[q].x += w.x; acc[q].y += w.y; acc[q].z += w.z; acc[q].w += w.w;
        }
    float4* o = (float4*)(WsJ + ((size_t)j * N_IN + n) * L_DIM);
#pragma unroll
    for (int q = 0; q < 4; ++q) o[q] = acc[q];
}

// ------- per-j GEMM over N (K=2304) with f32 WMMA, fused squash + U update -------
// mode 0: U = v (iter0, preScale=1/32); mode 1: U += v; mode 2: out = v (final)
__global__ __launch_bounds__(128) void gemm_squash(
    const float* __restrict__ Asrc, int jstride,
    const float* __restrict__ WsJ,
    float* __restrict__ U, float* __restrict__ out,
    float preScale, int mode) {
    __shared__ float Asub[16 * ASTR];   // [16 b][CH n]
    __shared__ float Bsub[CH * 16];     // [CH n][16 l]
    __shared__ float red[3 * 256];      // wave 1..3 partial accumulators

    const int j    = blockIdx.x;
    const int b0   = blockIdx.y * 16;
    const int tid  = threadIdx.x;
    const int lane = tid & 31;
    const int wave = tid >> 5;

    const float* Abase = Asrc + (size_t)j * jstride;                 // [b][n] rows
    const float* Bbase = WsJ + (size_t)j * ((size_t)N_IN * L_DIM);   // [n][l] rows

    const int row16 = lane & 15;          // A: M row / B,D: column l
    const int kb    = (lane >> 4) * 2;    // K sub-offset per lane half

    // per-thread staging coordinates (2 float4's per tile per thread)
    const int ar0 = tid >> 4, ac0 = (tid & 15) * 4;          // A: rows 0..7
    const int ar1 = ar0 + 8;                                  // A: rows 8..15

    v8f c = {};
    for (int n0 = 0; n0 < N_IN; n0 += CH) {
        // stage A tile: 16x64 floats as 2 x b128 per thread (coalesced rows)
        float4 a0 = *(const float4*)(Abase + (size_t)(b0 + ar0) * N_IN + n0 + ac0);
        float4 a1 = *(const float4*)(Abase + (size_t)(b0 + ar1) * N_IN + n0 + ac0);
        // stage B tile: 64x16 floats fully contiguous, 2 x b128 per thread
        float4 bq0 = *(const float4*)(Bbase + (size_t)n0 * L_DIM + tid * 4);
        float4 bq1 = *(const float4*)(Bbase + (size_t)n0 * L_DIM + (tid + 128) * 4);
        *(float4*)(&Asub[ar0 * ASTR + ac0]) = a0;
        *(float4*)(&Asub[ar1 * ASTR + ac0]) = a1;
        *(float4*)(&Bsub[tid * 4])          = bq0;
        *(float4*)(&Bsub[(tid + 128) * 4])  = bq1;
        __syncthreads();

        const int base = wave * 16;   // each wave owns a 16-n slice of the chunk
#pragma unroll
        for (int kk = 0; kk < 16; kk += 4) {
            int nl = base + kk;
            v2f a, b;
            a.x = Asub[row16 * ASTR + nl + kb];
            a.y = Asub[row16 * ASTR + nl + kb + 1];
            b.x = Bsub[(nl + kb) * 16 + row16];
            b.y = Bsub[(nl + kb + 1) * 16 + row16];
            c = __builtin_amdgcn_wmma_f32_16x16x4_f32(
                    false, a, false, b, (short)0, c, false, false);
        }
        __syncthreads();
    }

    // cross-wave reduction of the 16x16 accumulator (vectorized spill)
    if (wave > 0) {
        float* r = &red[(wave - 1) * 256 + lane * 8];
        *(float4*)(r)     = make_float4(c[0], c[1], c[2], c[3]);
        *(float4*)(r + 4) = make_float4(c[4], c[5], c[6], c[7]);
    }
    __syncthreads();

    if (wave == 0) {
#pragma unroll
        for (int w = 0; w < 3; ++w) {
            const float* r = &red[w * 256 + lane * 8];
            float4 p0 = *(const float4*)(r);
            float4 p1 = *(const float4*)(r + 4);
            c[0] += p0.x; c[1] += p0.y; c[2] += p0.z; c[3] += p0.w;
            c[4] += p1.x; c[5] += p1.y; c[6] += p1.z; c[7] += p1.w;
        }

        const int half = lane >> 4, col = lane & 15;
#pragma unroll
        for (int r = 0; r < 8; ++r) {
            float s = c[r] * preScale;
            float t = s * s;
#pragma unroll
            for (int m = 1; m < 16; m <<= 1)   // masks 1,2,4,8: stay in 16-lane half
                t += __shfl_xor(t, m, 32);
            float scale = t / (1.0f + t) * rsqrtf(t + EPSQ);
            float v = scale * s;
            int brow = b0 + r + half * 8;
            size_t o = (size_t)brow * JL + (size_t)j * L_DIM + col;
            if (mode == 0)      U[o] = v;
            else if (mode == 1) U[o] += v;
            else                out[o] = v;
        }
    }
}

// ------- routing coefficients: Mcoef[j][b][n] = xs * softmax_j(xs*<Ws[n,j],U[b,j]>) -------
__global__ __launch_bounds__(256) void coef_kernel(
    const float* __restrict__ xs, const float* __restrict__ WsJ,
    const float* __restrict__ U, float* __restrict__ Mcoef) {
    __shared__ float Uv[JL];
    const int b = blockIdx.y;
    const int n = blockIdx.x * 256 + threadIdx.x;
    if (threadIdx.x < 128) {
        float4 u = *(const float4*)(U + (size_t)b * JL + threadIdx.x * 4);
        *(float4*)(&Uv[threadIdx.x * 4]) = u;
    }
    __syncthreads();

    const float xsv = xs[(size_t)b * N_IN + n];
    float t[J_CAPS];
#pragma unroll 4
    for (int j = 0; j < J_CAPS; ++j) {
        const float4* p = (const float4*)(WsJ + ((size_t)j * N_IN + n) * L_DIM);
        const float* u = Uv + j * L_DIM;
        float acc = 0.f;
#pragma unroll
        for (int q = 0; q < 4; ++q) {
            float4 w = p[q];
            acc += w.x * u[q * 4 + 0] + w.y * u[q * 4 + 1] +
                   w.z * u[q * 4 + 2] + w.w * u[q * 4 + 3];
        }
        t[j] = xsv * acc;
    }
    float mx = t[0];
#pragma unroll
    for (int j = 1; j < J_CAPS; ++j) mx = fmaxf(mx, t[j]);
    float sum = 0.f;
#pragma unroll
    for (int j = 0; j < J_CAPS; ++j) { t[j] = __expf(t[j] - mx); sum += t[j]; }
    const float inv = xsv / sum;     // fold xs into M = xs * c
#pragma unroll
    for (int j = 0; j < J_CAPS; ++j)
        Mcoef[((size_t)j * B_SZ + b) * N_IN + n] = t[j] * inv;
}

extern "C" void kernel_launch(void* const* d_in, const int* in_sizes, int n_in,
                              void* d_out, int out_size, void* d_ws, size_t ws_size,
                              hipStream_t stream) {
    (void)in_sizes; (void)n_in; (void)out_size; (void)ws_size;
    const float* x = (const float*)d_in[0];
    const float* W = (const float*)d_in[1];
    float* out = (float*)d_out;

    float* ws  = (float*)d_ws;
    float* xs  = ws;                                        // 147456 floats
    float* WsJ = xs + (size_t)B_SZ * N_IN;                  // 1179648
    float* U   = WsJ + (size_t)J_CAPS * N_IN * L_DIM;       // 32768
    float* Mc  = U + (size_t)B_SZ * JL;                     // 4718592  (~24.3 MB total)

    prep_x<<<(B_SZ * N_IN) / 256, 256, 0, stream>>>(x, xs);
    prep_w<<<(N_IN * J_CAPS) / 256, 256, 0, stream>>>(W, WsJ);

    dim3 ggrid(J_CAPS, B_SZ / 16);
    dim3 cgrid(N_IN / 256, B_SZ);

    // iter 0: uniform c = 1/J  (pure GEMM path), U = v0
    gemm_squash<<<ggrid, 128, 0, stream>>>(xs, 0, WsJ, U, out, 1.0f / J_CAPS, 0);
    // iter 1
    coef_kernel<<<cgrid, 256, 0, stream>>>(xs, WsJ, U, Mc);
    gemm_squash<<<ggrid, 128, 0, stream>>>(Mc, B_SZ * N_IN, WsJ, U, out, 1.0f, 1);
    // iter 2 (final): write v to d_out
    coef_kernel<<<cgrid, 256, 0, stream>>>(xs, WsJ, U, Mc);
    gemm_squash<<<ggrid, 128, 0, stream>>>(Mc, B_SZ * N_IN, WsJ, U, out, 1.0f, 2);
}